// GraphTemporalRNNConv_48610439856736
// MI455X (gfx1250) — compile-verified
//
#include <hip/hip_runtime.h>
#include <hip/hip_bf16.h>
#include <math.h>

// Problem constants (match reference)
#define NN 5000
#define EE 160000
#define RR 8
#define DD 512
#define NBLK 8
#define DB 64
#define EP (EE + 32 * RR)       // padded permutation length (160256)
#define NCHUNK (EP / 32)        // 5008 chunks of 32 edges
#define MT32 ((NN + 31) / 32)   // 157 row tiles of 32

typedef __attribute__((ext_vector_type(2))) float v2f;
typedef __attribute__((ext_vector_type(8))) float v8f;

__device__ __forceinline__ v8f wmma4(v2f a, v2f b, v8f c) {
    // V_WMMA_F32_16X16X4_F32 : D = A(16x4) * B(4x16) + C
    return __builtin_amdgcn_wmma_f32_16x16x4_f32(false, a, false, b, (short)0, c, false, false);
}

__device__ __forceinline__ float sigmf_(float x) { return 1.0f / (1.0f + __expf(-x)); }

// ---------------------------------------------------------------- edge norms
__global__ void k_edge_norm(const int* __restrict__ src, const int* __restrict__ dst,
                            const float* __restrict__ et, const float* __restrict__ nlet,
                            float* __restrict__ nf, float* __restrict__ nr) {
    int e = blockIdx.x * blockDim.x + threadIdx.x;
    if (e >= EE) return;
    int s = src[e], d = dst[e];
    float dtf = et[e] - nlet[(size_t)d * (NN + 1) * 2 + (size_t)s * 2 + 0];
    float dtr = et[e] - nlet[(size_t)s * (NN + 1) * 2 + (size_t)d * 2 + 1];
    float tf = log1pf(fmaxf(dtf, 0.0f));
    float tr = log1pf(fmaxf(dtr, 0.0f));
    nf[e] = fminf(1.0f / fmaxf(tf, 1e-10f), 10.0f);
    nr[e] = fminf(1.0f / fmaxf(tr, 1e-10f), 10.0f);
}

// ------------------------------------------------- relation counting sort
__global__ void k_zero16(int* __restrict__ p) {
    if (threadIdx.x < 16) p[threadIdx.x] = 0;
}
__global__ void k_fill_neg(int* __restrict__ p, int n) {
    int i = blockIdx.x * blockDim.x + threadIdx.x;
    if (i < n) p[i] = -1;
}
__global__ void k_hist(const int* __restrict__ rel, int* __restrict__ cnt) {
    int e = blockIdx.x * blockDim.x + threadIdx.x;
    if (e < EE) atomicAdd(&cnt[rel[e]], 1);
}
__global__ void k_offsets(const int* __restrict__ cnt, int* __restrict__ off) {
    if (threadIdx.x == 0 && blockIdx.x == 0) {
        int acc = 0;
        for (int r = 0; r < RR; ++r) {
            off[r] = acc;
            acc += ((cnt[r] + 31) >> 5) << 5;   // pad group to multiple of 32
        }
    }
}
__global__ void k_scatter(const int* __restrict__ rel, int* __restrict__ cursor,
                          const int* __restrict__ off, int* __restrict__ permP) {
    int e = blockIdx.x * blockDim.x + threadIdx.x;
    if (e >= EE) return;
    int r = rel[e];
    int pos = atomicAdd(&cursor[r], 1);
    permP[off[r] + pos] = e;
}

// ----------------------------------------------------- weight pre-packing
// Pack Wt (K x Nc row-major, or its transpose source) into row-pair interleaved
// layout so a WMMA B-fragment (rows k,k+1 at one column) is one b64 load:
//   P[(k>>1)*2*Nc + n*2 + (k&1)] = Wt[k][n]
__global__ void k_pack(const float* __restrict__ S, float* __restrict__ P,
                       int K, int Nc, int batch, int transpose) {
    size_t total = (size_t)K * Nc * batch;
    size_t i = (size_t)blockIdx.x * blockDim.x + threadIdx.x;
    if (i >= total) return;
    int per = K * Nc;
    int b = (int)(i / per);
    int r = (int)(i - (size_t)b * per);
    int k = r / Nc, n = r - k * Nc;
    float v = transpose ? S[(size_t)b * per + (size_t)n * K + k]
                        : S[(size_t)b * per + (size_t)k * Nc + n];
    P[(size_t)b * per + (size_t)(k >> 1) * 2 * Nc + n * 2 + (k & 1)] = v;
}

// ------------------------------------------------------------ dense GEMM
// Y[M x Dout] = X[M x K] @ Wt[K x Dout] + bias (Wt given packed, ldw = Dout).
// Wave computes a 32x64 tile: 8 accumulators, 8 WMMAs per K-step of 4,
// fed by 2 A b64-loads + 4 packed-B b64-loads.
template <int SX>
__global__ void k_gemm(const float* __restrict__ X, int ldx,
                       const float* __restrict__ Wp, int ldw,
                       const float* __restrict__ bias,
                       float* __restrict__ Y, int ldy,
                       int M, int K, int Dout) {
    int wave = threadIdx.x >> 5;
    int lane = threadIdx.x & 31;
    int n0 = (blockIdx.y * 8 + wave) * 64;
    if (n0 >= Dout) return;                    // wave-uniform exit
    int m0 = blockIdx.x * 32;
    int lh = lane >> 4;
    int l15 = lane & 15;
    int r0 = m0 + l15;      if (r0 >= M) r0 = M - 1;
    int r1 = m0 + 16 + l15; if (r1 >= M) r1 = M - 1;
    const float* x0 = X + (size_t)r0 * ldx * SX;
    const float* x1 = X + (size_t)r1 * ldx * SX;
    const float* wb = Wp + (size_t)(n0 + l15) * 2;
    v8f c00 = {}, c01 = {}, c02 = {}, c03 = {};
    v8f c10 = {}, c11 = {}, c12 = {}, c13 = {};
    #pragma unroll 2
    for (int k = 0; k < K; k += 4) {
        int ka = k + lh * 2;
        v2f a0, a1;
        if (SX == 1) {
            a0 = *(const v2f*)(x0 + ka);
            a1 = *(const v2f*)(x1 + ka);
        } else {
            a0.x = x0[(size_t)ka * SX]; a0.y = x0[(size_t)(ka + 1) * SX];
            a1.x = x1[(size_t)ka * SX]; a1.y = x1[(size_t)(ka + 1) * SX];
        }
        const float* wp = wb + (size_t)(ka >> 1) * 2 * ldw;
        v2f b0 = *(const v2f*)(wp);
        v2f b1 = *(const v2f*)(wp + 32);
        v2f b2 = *(const v2f*)(wp + 64);
        v2f b3 = *(const v2f*)(wp + 96);
        c00 = wmma4(a0, b0, c00); c01 = wmma4(a0, b1, c01);
        c02 = wmma4(a0, b2, c02); c03 = wmma4(a0, b3, c03);
        c10 = wmma4(a1, b0, c10); c11 = wmma4(a1, b1, c11);
        c12 = wmma4(a1, b2, c12); c13 = wmma4(a1, b3, c13);
    }
    float bb0 = bias[n0 + l15];
    float bb1 = bias[n0 + 16 + l15];
    float bb2 = bias[n0 + 32 + l15];
    float bb3 = bias[n0 + 48 + l15];
    #pragma unroll
    for (int v = 0; v < 8; ++v) {
        int m = m0 + (lh << 3) + v;
        if (m < M) {
            float* yr = Y + (size_t)m * ldy + n0 + l15;
            yr[0] = c00[v] + bb0; yr[16] = c01[v] + bb1;
            yr[32] = c02[v] + bb2; yr[48] = c03[v] + bb3;
        }
        int m2 = m + 16;
        if (m2 < M) {
            float* yr = Y + (size_t)m2 * ldy + n0 + l15;
            yr[0] = c10[v] + bb0; yr[16] = c11[v] + bb1;
            yr[32] = c12[v] + bb2; yr[48] = c13[v] + bb3;
        }
    }
}

// --------------------------------------------- per-edge bdd message + scatter
// One workgroup per 32-edge chunk (same relation); wave w handles diag block b=w
// with a 32x64 WMMA tile against the packed 64x64 relation block.
// A rows pre-scaled by edge norm (padding rows scale 0 -> exact zero contribution).
__global__ void k_edge_msg(const float* __restrict__ h, float* __restrict__ agg,
                           const float* __restrict__ norm,
                           const int* __restrict__ srcA, const int* __restrict__ dstA,
                           const int* __restrict__ rel, const int* __restrict__ permP,
                           const float* __restrict__ WrelP) {
    int wave = threadIdx.x >> 5;               // diagonal block index b
    int lane = threadIdx.x & 31;
    int base = blockIdx.x * 32;
    int e0 = permP[base];
    if (e0 < 0) return;                        // unused chunk (uniform)
    int r = rel[e0];
    const float* Wb = WrelP + ((size_t)(r * NBLK + wave)) * DB * DB; // packed block
    int lh = lane >> 4, l15 = lane & 15;
    int pa = permP[base + l15];
    int pb = permP[base + 16 + l15];
    int ea = (pa < 0) ? e0 : pa;
    int eb = (pb < 0) ? e0 : pb;
    float sa = (pa < 0) ? 0.0f : norm[ea];
    float sb = (pb < 0) ? 0.0f : norm[eb];
    const float* x0 = h + (size_t)srcA[ea] * DD + wave * DB;
    const float* x1 = h + (size_t)srcA[eb] * DD + wave * DB;
    const float* wbp = Wb + (size_t)l15 * 2;
    v8f c00 = {}, c01 = {}, c02 = {}, c03 = {};
    v8f c10 = {}, c11 = {}, c12 = {}, c13 = {};
    #pragma unroll 4
    for (int k = 0; k < DB; k += 4) {
        int ka = k + lh * 2;
        v2f a0 = *(const v2f*)(x0 + ka); a0.x *= sa; a0.y *= sa;
        v2f a1 = *(const v2f*)(x1 + ka); a1.x *= sb; a1.y *= sb;
        const float* wp = wbp + (size_t)(ka >> 1) * 2 * DB;
        v2f b0 = *(const v2f*)(wp);
        v2f b1 = *(const v2f*)(wp + 32);
        v2f b2 = *(const v2f*)(wp + 64);
        v2f b3 = *(const v2f*)(wp + 96);
        c00 = wmma4(a0, b0, c00); c01 = wmma4(a0, b1, c01);
        c02 = wmma4(a0, b2, c02); c03 = wmma4(a0, b3, c03);
        c10 = wmma4(a1, b0, c10); c11 = wmma4(a1, b1, c11);
        c12 = wmma4(a1, b2, c12); c13 = wmma4(a1, b3, c13);
    }
    #pragma unroll
    for (int v = 0; v < 8; ++v) {
        int m = (lh << 3) + v;
        int pm0 = permP[base + m];
        int pm1 = permP[base + 16 + m];
        int em0 = (pm0 < 0) ? e0 : pm0;        // padded rows carry exact zeros
        int em1 = (pm1 < 0) ? e0 : pm1;
        float* ar0 = agg + (size_t)dstA[em0] * DD + wave * DB + l15;
        float* ar1 = agg + (size_t)dstA[em1] * DD + wave * DB + l15;
        atomicAdd(ar0 + 0,  c00[v]); atomicAdd(ar0 + 16, c01[v]);
        atomicAdd(ar0 + 32, c02[v]); atomicAdd(ar0 + 48, c03[v]);
        atomicAdd(ar1 + 0,  c10[v]); atomicAdd(ar1 + 16, c11[v]);
        atomicAdd(ar1 + 32, c12[v]); atomicAdd(ar1 + 48, c13[v]);
    }
}

// ------------------------------------------------------------- elementwise
__global__ void k_relu(float* __restrict__ x, int n) {
    int i = blockIdx.x * blockDim.x + threadIdx.x;
    if (i < n) x[i] = fmaxf(x[i], 0.0f);
}
__global__ void k_gru(const float* __restrict__ gi, const float* __restrict__ gh,
                      const float* __restrict__ dyn, float* __restrict__ out, int dir) {
    int i = blockIdx.x * blockDim.x + threadIdx.x;
    if (i >= NN * DD) return;
    int n = i >> 9, d = i & (DD - 1);
    const float* g = gi + (size_t)n * 3 * DD;
    const float* hh = gh + (size_t)n * 3 * DD;
    float r  = sigmf_(g[d] + hh[d]);
    float z  = sigmf_(g[DD + d] + hh[DD + d]);
    float ng = tanhf(g[2 * DD + d] + r * hh[2 * DD + d]);
    float hp = dyn[((size_t)n * DD + d) * 2 + dir];
    out[((size_t)n * DD + d) * 2 + dir] = (1.0f - z) * ng + z * hp;
}

// ----------------------------------------------------------------- launch
extern "C" void kernel_launch(void* const* d_in, const int* in_sizes, int n_in,
                              void* d_out, int out_size, void* d_ws, size_t ws_size,
                              hipStream_t stream) {
    (void)in_sizes; (void)n_in; (void)out_size; (void)ws_size;
    const int*   src        = (const int*)d_in[0];
    const int*   dst        = (const int*)d_in[1];
    const int*   rel        = (const int*)d_in[2];
    const float* edge_time  = (const float*)d_in[3];
    const float* nlet       = (const float*)d_in[4];
    const float* static_emb = (const float*)d_in[5];
    const float* dyn        = (const float*)d_in[6];
    const float* Wrel1      = (const float*)d_in[7];
    const float* Wself1     = (const float*)d_in[8];
    const float* b1         = (const float*)d_in[9];
    const float* Wrel2      = (const float*)d_in[10];
    const float* Wself2     = (const float*)d_in[11];
    const float* b2         = (const float*)d_in[12];
    const float* Wih        = (const float*)d_in[13];
    const float* Whh        = (const float*)d_in[14];
    const float* bih        = (const float*)d_in[15];
    const float* bhh        = (const float*)d_in[16];
    float* out = (float*)d_out;

    char* ws = (char*)d_ws;
    size_t o = 0;
    auto take = [&](size_t bytes) -> char* {
        char* p = ws + o; o = (o + bytes + 255) & ~(size_t)255; return p;
    };
    float* norm_f  = (float*)take((size_t)EE * 4);
    float* norm_r  = (float*)take((size_t)EE * 4);
    int*   permP   = (int*)  take((size_t)EP * 4);
    int*   cnt     = (int*)  take(64);             // cnt[8] + cursor[8]
    int*   cursor  = cnt + 8;
    int*   off     = (int*)  take(64);
    float* WselfP1 = (float*)take((size_t)DD * DD * 4);
    float* WselfP2 = (float*)take((size_t)DD * DD * 4);
    float* WihP    = (float*)take((size_t)DD * 3 * DD * 4);
    float* WhhP    = (float*)take((size_t)DD * 3 * DD * 4);
    float* WrelP1  = (float*)take((size_t)RR * NBLK * DB * DB * 4);
    float* WrelP2  = (float*)take((size_t)RR * NBLK * DB * DB * 4);
    float* h1      = (float*)take((size_t)NN * DD * 4);
    float* h2      = (float*)take((size_t)NN * DD * 4);
    float* gi      = (float*)take((size_t)NN * 3 * DD * 4);
    float* gh      = (float*)take((size_t)NN * 3 * DD * 4);

    const int T = 256;
    // pack all weights into WMMA-B row-pair-interleaved layout (once per launch)
    k_pack<<<(DD * DD + T - 1) / T, T, 0, stream>>>(Wself1, WselfP1, DD, DD, 1, 0);
    k_pack<<<(DD * DD + T - 1) / T, T, 0, stream>>>(Wself2, WselfP2, DD, DD, 1, 0);
    k_pack<<<(3 * DD * DD + T - 1) / T, T, 0, stream>>>(Wih, WihP, DD, 3 * DD, 1, 1);
    k_pack<<<(3 * DD * DD + T - 1) / T, T, 0, stream>>>(Whh, WhhP, DD, 3 * DD, 1, 1);
    k_pack<<<(RR * NBLK * DB * DB + T - 1) / T, T, 0, stream>>>(Wrel1, WrelP1, DB, DB, RR * NBLK, 0);
    k_pack<<<(RR * NBLK * DB * DB + T - 1) / T, T, 0, stream>>>(Wrel2, WrelP2, DB, DB, RR * NBLK, 0);
    // edge norms
    k_edge_norm<<<(EE + T - 1) / T, T, 0, stream>>>(src, dst, edge_time, nlet, norm_f, norm_r);
    // relation counting sort into 32-padded groups
    k_zero16<<<1, 32, 0, stream>>>(cnt);
    k_fill_neg<<<(EP + T - 1) / T, T, 0, stream>>>(permP, EP);
    k_hist<<<(EE + T - 1) / T, T, 0, stream>>>(rel, cnt);
    k_offsets<<<1, 1, 0, stream>>>(cnt, off);
    k_scatter<<<(EE + T - 1) / T, T, 0, stream>>>(rel, cursor, off, permP);

    for (int dir = 0; dir < 2; ++dir) {
        const int* sA = dir ? dst : src;
        const int* dA = dir ? src : dst;
        const float* nrm = dir ? norm_r : norm_f;
        // layer 1: h1 = relu(static_emb@Wself1 + b1 + scatter(bdd msgs))
        k_gemm<1><<<dim3(MT32, 1), T, 0, stream>>>(static_emb, DD, WselfP1, DD, b1, h1, DD, NN, DD, DD);
        k_edge_msg<<<NCHUNK, T, 0, stream>>>(static_emb, h1, nrm, sA, dA, rel, permP, WrelP1);
        k_relu<<<(NN * DD + T - 1) / T, T, 0, stream>>>(h1, NN * DD);
        // layer 2
        k_gemm<1><<<dim3(MT32, 1), T, 0, stream>>>(h1, DD, WselfP2, DD, b2, h2, DD, NN, DD, DD);
        k_edge_msg<<<NCHUNK, T, 0, stream>>>(h1, h2, nrm, sA, dA, rel, permP, WrelP2);
        k_relu<<<(NN * DD + T - 1) / T, T, 0, stream>>>(h2, NN * DD);
        // GRU: gi = h2@Wih^T + bih ; gh = dyn[:,:,dir]@Whh^T + bhh
        k_gemm<1><<<dim3(MT32, 3), T, 0, stream>>>(h2, DD, WihP, 3 * DD, bih, gi, 3 * DD, NN, DD, 3 * DD);
        k_gemm<2><<<dim3(MT32, 3), T, 0, stream>>>(dyn + dir, DD, WhhP, 3 * DD, bhh, gh, 3 * DD, NN, DD, 3 * DD);
        k_gru<<<(NN * DD + T - 1) / T, T, 0, stream>>>(gi, gh, dyn, out, dir);
    }
}